// CVGAE_63213328662976
// MI455X (gfx1250) — compile-verified
//
#include <hip/hip_runtime.h>
#include <hip/hip_bf16.h>

#define N_ 8192
#define F_ 1024
#define H_ 64
#define D_ 32
#define K_ 16
#define E_ (N_ * 32)

typedef float v2f __attribute__((ext_vector_type(2)));
typedef float v8f __attribute__((ext_vector_type(8)));

__device__ __forceinline__ v8f wmma4(v2f a, v2f b, v8f c) {
    // D = A(16x4, f32) * B(4x16, f32) + C(16x16, f32)
    return __builtin_amdgcn_wmma_f32_16x16x4_f32(false, a, false, b, (short)0, c,
                                                 false, false);
}

// ---------------- zero / elementwise ----------------

__global__ void zero_kernel(float4* __restrict__ p) {
    int i = blockIdx.x * blockDim.x + threadIdx.x;
    p[i] = make_float4(0.f, 0.f, 0.f, 0.f);
}

__global__ void relu_kernel(float4* __restrict__ h) {
    int i = blockIdx.x * blockDim.x + threadIdx.x;
    float4 v = h[i];
    v.x = fmaxf(v.x, 0.f);
    v.y = fmaxf(v.y, 0.f);
    v.z = fmaxf(v.z, 0.f);
    v.w = fmaxf(v.w, 0.f);
    h[i] = v;
}

__global__ void z_kernel(const float* __restrict__ mean, const float* __restrict__ ls,
                         const float* __restrict__ eps, float* __restrict__ z) {
    int i = blockIdx.x * blockDim.x + threadIdx.x;
    z[i] = mean[i] + eps[i] * __expf(0.5f * ls[i]);
}

// ---------------- GEMM: tmp0 = x @ w0  ([8192,1024] x [1024,64]) ----------------
// one wave per 16-row strip, computing 16x64 (A fragment reused across 4 col tiles)

__global__ void gemm_xw0_kernel(const float* __restrict__ x,
                                const float* __restrict__ w0,
                                float* __restrict__ out) {
    const int lane = threadIdx.x & 31;
    const int half = lane >> 4;       // 0: K=0,1  1: K=2,3
    const int l    = lane & 15;
    const int strip = (blockIdx.x * blockDim.x + threadIdx.x) >> 5;  // 0..511
    const int rowBase = strip * 16;

    const float* xr = x + (size_t)(rowBase + l) * F_ + 2 * half;
    v8f acc0 = {}, acc1 = {}, acc2 = {}, acc3 = {};
    for (int k = 0; k < F_; k += 4) {
        v2f a;
        a.x = xr[k];
        a.y = xr[k + 1];
        const float* b0 = w0 + (size_t)(k + 2 * half) * H_ + l;
        v2f b;
        b.x = b0[0];  b.y = b0[H_];      acc0 = wmma4(a, b, acc0);
        b.x = b0[16]; b.y = b0[H_ + 16]; acc1 = wmma4(a, b, acc1);
        b.x = b0[32]; b.y = b0[H_ + 32]; acc2 = wmma4(a, b, acc2);
        b.x = b0[48]; b.y = b0[H_ + 48]; acc3 = wmma4(a, b, acc3);
    }
    float* o = out + (size_t)(rowBase + half * 8) * H_ + l;
#pragma unroll
    for (int r = 0; r < 8; ++r) {
        o[r * H_ + 0]  = acc0[r];
        o[r * H_ + 16] = acc1[r];
        o[r * H_ + 32] = acc2[r];
        o[r * H_ + 48] = acc3[r];
    }
}

// ---------------- GEMM: {mean_pre, ls_pre} = h0 @ {w_mu, w_ls} ----------------

__global__ void gemm_h_kernel(const float* __restrict__ h,
                              const float* __restrict__ wmu,
                              const float* __restrict__ wls,
                              float* __restrict__ mean_pre,
                              float* __restrict__ ls_pre) {
    const int lane = threadIdx.x & 31;
    const int half = lane >> 4;
    const int l    = lane & 15;
    const int strip = (blockIdx.x * blockDim.x + threadIdx.x) >> 5;  // 0..511
    const int rowBase = strip * 16;

    const float* hr = h + (size_t)(rowBase + l) * H_ + 2 * half;
    v8f am0 = {}, am1 = {}, al0 = {}, al1 = {};
#pragma unroll
    for (int k = 0; k < H_; k += 4) {
        v2f a;
        a.x = hr[k];
        a.y = hr[k + 1];
        const float* bm = wmu + (size_t)(k + 2 * half) * D_ + l;
        const float* bl = wls + (size_t)(k + 2 * half) * D_ + l;
        v2f b;
        b.x = bm[0];  b.y = bm[D_];      am0 = wmma4(a, b, am0);
        b.x = bm[16]; b.y = bm[D_ + 16]; am1 = wmma4(a, b, am1);
        b.x = bl[0];  b.y = bl[D_];      al0 = wmma4(a, b, al0);
        b.x = bl[16]; b.y = bl[D_ + 16]; al1 = wmma4(a, b, al1);
    }
    float* om = mean_pre + (size_t)(rowBase + half * 8) * D_ + l;
    float* ol = ls_pre   + (size_t)(rowBase + half * 8) * D_ + l;
#pragma unroll
    for (int r = 0; r < 8; ++r) {
        om[r * D_ + 0]  = am0[r];
        om[r * D_ + 16] = am1[r];
        ol[r * D_ + 0]  = al0[r];
        ol[r * D_ + 16] = al1[r];
    }
}

// ---------------- SpMM (edge-parallel, f32 atomics) ----------------

__global__ void spmm_h_kernel(const int* __restrict__ src, const int* __restrict__ dst,
                              const float* __restrict__ w, const float* __restrict__ tin,
                              float* __restrict__ hout) {
    unsigned tid = blockIdx.x * blockDim.x + threadIdx.x;  // E_*64 threads exactly
    int e = tid >> 6;
    int f = tid & 63;
    float v = w[e] * tin[(size_t)src[e] * H_ + f];
    atomicAdd(&hout[(size_t)dst[e] * H_ + f], v);
}

__global__ void spmm_md_kernel(const int* __restrict__ src, const int* __restrict__ dst,
                               const float* __restrict__ w,
                               const float* __restrict__ mpre, const float* __restrict__ lpre,
                               float* __restrict__ mout, float* __restrict__ lout) {
    unsigned tid = blockIdx.x * blockDim.x + threadIdx.x;  // E_*64 threads exactly
    int e = tid >> 6;
    int t = tid & 63;
    int f = t & 31;
    int s = src[e], d = dst[e];
    float we = w[e];
    if (t < 32)
        atomicAdd(&mout[(size_t)d * D_ + f], we * mpre[(size_t)s * D_ + f]);
    else
        atomicAdd(&lout[(size_t)d * D_ + f], we * lpre[(size_t)s * D_ + f]);
}

// ---------------- A_pred = sigmoid(z @ z^T)  (dominant: 256 MB NT stores) ----------------
// one wave per 16x16 tile; K=32 -> 8 WMMAs, fully unrolled

__global__ void apred_kernel(const float* __restrict__ z, float* __restrict__ out) {
    const int lane = threadIdx.x & 31;
    const int half = lane >> 4;
    const int l    = lane & 15;
    const int gw = (blockIdx.x * blockDim.x + threadIdx.x) >> 5;  // 0..262143
    const int tileM = gw >> 9;
    const int tileN = gw & 511;
    const int rowBase = tileM * 16;
    const int colBase = tileN * 16;

    const float* zr = z + (size_t)(rowBase + l) * D_ + 2 * half;
    const float* zc = z + (size_t)(colBase + l) * D_ + 2 * half;  // B = z^T
    v8f acc = {};
#pragma unroll
    for (int k = 0; k < D_; k += 4) {
        v2f a; a.x = zr[k]; a.y = zr[k + 1];
        v2f b; b.x = zc[k]; b.y = zc[k + 1];
        acc = wmma4(a, b, acc);
    }
    float* o = out + (size_t)(rowBase + half * 8) * N_ + colBase + l;
#pragma unroll
    for (int r = 0; r < 8; ++r) {
        float v = acc[r];
        float s = 1.0f / (1.0f + __expf(-v));
        __builtin_nontemporal_store(s, o + (size_t)r * N_);  // bypass caches: 256 MB stream
    }
}

// ---------------- Student-t cluster assignment (ALPHA=1 -> numerator = 1/(1+d2)) -------

__global__ void cluster_kernel(const float* __restrict__ z, const float* __restrict__ c,
                               float* __restrict__ p) {
    int i = blockIdx.x * blockDim.x + threadIdx.x;  // 0..8191
    float zi[D_];
#pragma unroll
    for (int d = 0; d < D_; ++d) zi[d] = z[(size_t)i * D_ + d];
    float num[K_];
    float s = 0.f;
#pragma unroll
    for (int k = 0; k < K_; ++k) {
        float d2 = 0.f;
#pragma unroll
        for (int d = 0; d < D_; ++d) {
            float df = zi[d] - c[k * D_ + d];
            d2 = fmaf(df, df, d2);
        }
        float nu = 1.0f / (1.0f + d2);
        num[k] = nu;
        s += nu;
    }
    float inv = 1.0f / s;
#pragma unroll
    for (int k = 0; k < K_; ++k) p[(size_t)i * K_ + k] = num[k] * inv;
}

// ---------------- driver ----------------

extern "C" void kernel_launch(void* const* d_in, const int* in_sizes, int n_in,
                              void* d_out, int out_size, void* d_ws, size_t ws_size,
                              hipStream_t stream) {
    const float* x    = (const float*)d_in[0];
    const int*   esrc = (const int*)d_in[1];
    const int*   edst = (const int*)d_in[2];
    const float* ew   = (const float*)d_in[3];
    const float* eps  = (const float*)d_in[4];
    const float* w0   = (const float*)d_in[5];
    const float* wmu  = (const float*)d_in[6];
    const float* wls  = (const float*)d_in[7];
    const float* cent = (const float*)d_in[8];

    float* ws   = (float*)d_ws;
    float* tmp0 = ws;                           // N*H
    float* h0   = tmp0 + (size_t)N_ * H_;       // N*H
    float* mpre = h0 + (size_t)N_ * H_;         // N*D
    float* lpre = mpre + (size_t)N_ * D_;       // N*D
    float* mean = lpre + (size_t)N_ * D_;       // N*D
    float* lsg  = mean + (size_t)N_ * D_;       // N*D
    float* z    = lsg + (size_t)N_ * D_;        // N*D   (total ~9 MB)

    float* Apred = (float*)d_out;
    float* p     = Apred + (size_t)N_ * N_;

    // zero the atomic accumulators (every call; deterministic)
    zero_kernel<<<(N_ * H_ / 4) / 256, 256, 0, stream>>>((float4*)h0);
    zero_kernel<<<(N_ * D_ / 4) / 256, 256, 0, stream>>>((float4*)mean);
    zero_kernel<<<(N_ * D_ / 4) / 256, 256, 0, stream>>>((float4*)lsg);

    // tmp0 = x @ w0          (512 waves, 16x64 tile each)
    gemm_xw0_kernel<<<(N_ / 16) * 32 / 256, 256, 0, stream>>>(x, w0, tmp0);
    // h0 = spmm(tmp0), then relu
    spmm_h_kernel<<<(unsigned)((size_t)E_ * 64 / 256), 256, 0, stream>>>(esrc, edst, ew,
                                                                         tmp0, h0);
    relu_kernel<<<(N_ * H_ / 4) / 256, 256, 0, stream>>>((float4*)h0);
    // mean_pre / ls_pre = h0 @ {w_mu, w_ls}
    gemm_h_kernel<<<(N_ / 16) * 32 / 256, 256, 0, stream>>>(h0, wmu, wls, mpre, lpre);
    // mean / logsigma2 = spmm(...)
    spmm_md_kernel<<<(unsigned)((size_t)E_ * 64 / 256), 256, 0, stream>>>(esrc, edst, ew,
                                                                          mpre, lpre, mean, lsg);
    // z = mean + eps * exp(0.5 * logsigma2)
    z_kernel<<<(N_ * D_) / 256, 256, 0, stream>>>(mean, lsg, eps, z);
    // A_pred = sigmoid(z @ z^T)   (262144 tiles, 8 waves/block)
    apred_kernel<<<(N_ / 16) * (N_ / 16) * 32 / 256, 256, 0, stream>>>(z, Apred);
    // p = Student-t soft assignment
    cluster_kernel<<<N_ / 256, 256, 0, stream>>>(z, cent, p);
}